// VolumetricCELoss_32177894981935
// MI455X (gfx1250) — compile-verified
//
#include <hip/hip_runtime.h>
#include <hip/hip_bf16.h>
#include <stdint.h>

// Problem constants (match reference setup_inputs)
#define NVOX     64
#define V3       (NVOX * NVOX * NVOX)   // 262144 voxels per slice
#define B_       8
#define J_       17
#define NSLICE   (B_ * J_)              // 136 (b,j) slices
#define BOXR     1000.0f
#define BETA     0.01f
#define EPS_     1e-6f
#define LOG2E    1.4426950408889634f

// Streaming config: 1024 threads x float4 = 16 KB per stage, 64 stages/slice,
// 4-deep async double(quad)-buffer in LDS (64 KB of the 320 KB WGP pool).
#define BLOCK       1024
#define VEC         4
#define STAGE_ELEMS (BLOCK * VEC)       // 4096 floats = 16 KB
#define NSTAGES     (V3 / STAGE_ELEMS)  // 64
#define NBUF        4

// CDNA5 async copy: global -> LDS, 16 B per lane, tracked by ASYNCcnt.
// GV addressing mode: per-lane 64-bit global address, LDS byte address in vdst VGPR.
__device__ __forceinline__ void async_copy_b128(uint32_t lds_byte_addr,
                                                const void* gaddr) {
  asm volatile("global_load_async_to_lds_b128 %0, %1, off"
               :
               : "v"(lds_byte_addr), "v"((uint64_t)(uintptr_t)gaddr)
               : "memory");
}

__device__ __forceinline__ uint32_t lds_addr_of(const void* p) {
  // Low 32 bits of a generic pointer into LDS == workgroup-relative LDS byte
  // address (flat LDS aperture = {SHARED_BASE, offset[31:0]}).
  return (uint32_t)(uintptr_t)p;
}

// Merge two online-softmax partials (base-2 domain): (m,s) <- (m,s) + (mo,so)
__device__ __forceinline__ void merge(float& m, float& s, float mo, float so) {
  float Mn = fmaxf(m, mo);
  s = s * exp2f(m - Mn) + so * exp2f(mo - Mn);
  m = Mn;
}

__global__ __launch_bounds__(BLOCK) void
VolumetricCELoss_slice_kernel(const float* __restrict__ vol,
                              const float* __restrict__ label,
                              const float* __restrict__ center,
                              float* __restrict__ partial) {
  __shared__ float4 sh[NBUF][BLOCK];   // 64 KB async staging buffers
  __shared__ float  smax[32];
  __shared__ float  ssum[32];

  const int tid   = threadIdx.x;
  const int slice = blockIdx.x;                 // one workgroup per (b,j)
  const float* sb = vol + (size_t)slice * V3;   // slice base (uniform -> SGPRs)

  // ---- Prologue: prime the async pipeline with stages 0..NBUF-2 ----
#pragma unroll
  for (int p = 0; p < NBUF - 1; ++p) {
    async_copy_b128(lds_addr_of(&sh[p][tid]),
                    sb + (size_t)p * STAGE_ELEMS + tid * VEC);
  }

  // ---- Main stream: online softmax in base-2 (exp2f == v_exp_f32) ----
  float m = -INFINITY, s = 0.0f;
  for (int k = 0; k < NSTAGES; ++k) {
    if (k + (NBUF - 1) < NSTAGES) {
      const int kn = k + (NBUF - 1);
      async_copy_b128(lds_addr_of(&sh[kn & (NBUF - 1)][tid]),
                      sb + (size_t)kn * STAGE_ELEMS + tid * VEC);
      // 4 ops in flight (k..k+3); ASYNCcnt completes in order -> <=3 means
      // stage k has landed in LDS for this wave's lanes.
      asm volatile("s_wait_asynccnt 0x3" ::: "memory");
    } else {
      asm volatile("s_wait_asynccnt 0x0" ::: "memory");
    }

    const float4 v = sh[k & (NBUF - 1)][tid];   // ds_load_b128, own lanes only
    const float y0 = v.x * LOG2E, y1 = v.y * LOG2E;
    const float y2 = v.z * LOG2E, y3 = v.w * LOG2E;
    const float m4 = fmaxf(fmaxf(y0, y1), fmaxf(y2, y3));
    const float Mn = fmaxf(m, m4);
    s = s * exp2f(m - Mn)
        + exp2f(y0 - Mn) + exp2f(y1 - Mn) + exp2f(y2 - Mn) + exp2f(y3 - Mn);
    m = Mn;
  }

  // ---- Reduce 1024 partials: wave32 shuffle tree, then 32 waves via LDS ----
#pragma unroll
  for (int off = 16; off >= 1; off >>= 1)
    merge(m, s, __shfl_xor(m, off, 32), __shfl_xor(s, off, 32));

  const int wave = tid >> 5, lane = tid & 31;
  if (lane == 0) { smax[wave] = m; ssum[wave] = s; }
  __syncthreads();

  if (wave == 0) {
    m = smax[lane];
    s = ssum[lane];                 // 32 waves == exactly one wave32 of work
#pragma unroll
    for (int off = 16; off >= 1; off >>= 1)
      merge(m, s, __shfl_xor(m, off, 32), __shfl_xor(s, off, 32));

    if (lane == 0) {
      // GT voxel index for this (b,j)
      const int b = slice / J_, j = slice % J_;
      int id[3];
#pragma unroll
      for (int d = 0; d < 3; ++d) {
        const float nrm = (label[(b * J_ + j) * 3 + d] - center[b * 3 + d]) / BOXR;
        int ii = (int)floorf((nrm + 1.0f) * 0.5f * (float)(NVOX - 1));
        id[d] = min(max(ii, 0), NVOX - 1);
      }
      const int flat = (id[0] * NVOX + id[1]) * NVOX + id[2];
      const float y      = sb[flat] * LOG2E;       // single scalar load (L2 hit)
      const float picked = exp2f(y - m) / s;       // == softmax(x)[flat]
      partial[slice]     = -logf(picked + EPS_);
    }
  }
}

// Deterministic fixed-order tree sum of the 136 per-slice terms.
__global__ __launch_bounds__(256) void
VolumetricCELoss_finalize_kernel(const float* __restrict__ partial,
                                 float* __restrict__ out) {
  __shared__ float red[256];
  const int t = threadIdx.x;
  red[t] = (t < NSLICE) ? partial[t] : 0.0f;
  __syncthreads();
#pragma unroll
  for (int off = 128; off >= 1; off >>= 1) {
    if (t < off) red[t] += red[t + off];
    __syncthreads();
  }
  if (t == 0) out[0] = BETA * red[0] / (float)NSLICE;
}

extern "C" void kernel_launch(void* const* d_in, const int* in_sizes, int n_in,
                              void* d_out, int out_size, void* d_ws, size_t ws_size,
                              hipStream_t stream) {
  const float* vol    = (const float*)d_in[0];  // [8,17,64,64,64] f32
  const float* label  = (const float*)d_in[1];  // [8,17,3] f32
  const float* center = (const float*)d_in[2];  // [8,3] f32
  float*       part   = (float*)d_ws;           // 136 floats of scratch
  float*       out    = (float*)d_out;          // scalar loss

  VolumetricCELoss_slice_kernel<<<NSLICE, BLOCK, 0, stream>>>(vol, label, center, part);
  VolumetricCELoss_finalize_kernel<<<1, 256, 0, stream>>>(part, out);
}